// AttnProcessor_LoRA_Capture_67516885893750
// MI455X (gfx1250) — compile-verified
//
#include <hip/hip_runtime.h>
#include <hip/hip_bf16.h>
#include <math.h>

// ---------------------------------------------------------------------------
// CDNA5 (gfx1250) fp32 WMMA building blocks: V_WMMA_F32_16X16X4_F32
//   A: 16x4 f32  -> v2f  (lane lo holds row m0+lo; k pair = k0 + 2*hi)
//   B: 4x16 f32  -> v2f  (lane lo holds col n0+lo; k pair = k0 + 2*hi)
//   C/D: 16x16 f32 -> v8f (VGPR r: rows r (lanes 0-15) and r+8 (lanes 16-31))
// ---------------------------------------------------------------------------
typedef float v2f __attribute__((ext_vector_type(2)));
typedef float v8f __attribute__((ext_vector_type(8)));
typedef unsigned int u32x4 __attribute__((ext_vector_type(4)));
typedef int i32x4 __attribute__((ext_vector_type(4)));
typedef int i32x8 __attribute__((ext_vector_type(8)));

#define WMMA_F32X4(a, b, c) \
    __builtin_amdgcn_wmma_f32_16x16x4_f32(false, (a), false, (b), (short)0, (c), false, false)

#define HEADS    20
#define HEAD_DIM 64
#define INNER    1280
#define SLEN     77
#define SPAD     80
#define KVSTRIDE 66   // LDS row stride in floats: 64 data + 2 pad dwords
                      // -> bank = (2n + k) % 64 conflict-free, 8B-aligned pairs

#if defined(__has_builtin)
#if __has_builtin(__builtin_amdgcn_tensor_load_to_lds) && __has_builtin(__builtin_amdgcn_s_wait_tensorcnt)
#define USE_TDM 1
#endif
#endif
#ifndef USE_TDM
#define USE_TDM 0
#endif

// ---------------------------------------------------------------------------
// Generic GEMM:  C[M,N] = alpha * A[M,K] @ Bw[N,K]^T  (+ C  if accumulate)
//                                                    (+ bias[n] if bias)
// Requirements: N % 64 == 0, K % 32 == 0, K >= 32.
// M arbitrary: A row index is CLAMPED (garbage rows only feed store-guarded
// output rows), so the K loop has no predication -> no exec-mask churn.
// Block = 128 threads (4 waves); each wave computes a 16(M) x 64(N) strip.
// K loop is double-buffered in 16-K chunks: 20 independent v2f loads per
// chunk are staged into distinct registers while the previous chunk's 16
// WMMAs execute, keeping loads in flight instead of s_wait_loadcnt 0 stalls.
// ---------------------------------------------------------------------------
__launch_bounds__(128)
__global__ void gemm_wmma_f32(const float* __restrict__ A,
                              const float* __restrict__ Bw,
                              float* __restrict__ C,
                              const float* __restrict__ bias,
                              int M, int N, int K,
                              float alpha, int accumulate)
{
    const int tid  = threadIdx.x;
    const int w    = tid >> 5;
    const int lane = tid & 31;
    const int lo   = lane & 15;
    const int hi   = lane >> 4;
    const int koff = 2 * hi;

    const int mtile = blockIdx.y * 64 + w * 16;
    const int nbase = blockIdx.x * 64;

    // Clamp instead of predicate: OOB rows produce garbage that is never stored.
    const int arow = mtile + lo;
    const int aclp = (arow < M) ? arow : (M - 1);
    const float* Ap  = A + (size_t)aclp * K + koff;
    const float* Bp0 = Bw + (size_t)(nbase + lo) * K + koff;
    const float* Bp1 = Bp0 + (size_t)16 * K;
    const float* Bp2 = Bp1 + (size_t)16 * K;
    const float* Bp3 = Bp2 + (size_t)16 * K;

    v8f c0 = (v8f)0.0f, c1 = (v8f)0.0f, c2 = (v8f)0.0f, c3 = (v8f)0.0f;

#define LOAD_CHUNK(aF, bF, kc_)                                   \
    _Pragma("unroll")                                             \
    for (int u = 0; u < 4; ++u) {                                 \
        aF[u]    = *(const v2f*)(Ap  + (kc_) + 4 * u);            \
        bF[u][0] = *(const v2f*)(Bp0 + (kc_) + 4 * u);            \
        bF[u][1] = *(const v2f*)(Bp1 + (kc_) + 4 * u);            \
        bF[u][2] = *(const v2f*)(Bp2 + (kc_) + 4 * u);            \
        bF[u][3] = *(const v2f*)(Bp3 + (kc_) + 4 * u);            \
    }

#define COMP_CHUNK(aF, bF)                                        \
    _Pragma("unroll")                                             \
    for (int u = 0; u < 4; ++u) {                                 \
        c0 = WMMA_F32X4(aF[u], bF[u][0], c0);                     \
        c1 = WMMA_F32X4(aF[u], bF[u][1], c1);                     \
        c2 = WMMA_F32X4(aF[u], bF[u][2], c2);                     \
        c3 = WMMA_F32X4(aF[u], bF[u][3], c3);                     \
    }

    v2f aA[4], bA[4][4];   // chunk buffer A (16 K-values)
    v2f aB[4], bB[4][4];   // chunk buffer B (16 K-values)

    LOAD_CHUNK(aA, bA, 0);
#pragma unroll 1
    for (int kc = 0; kc < K; kc += 32) {
        LOAD_CHUNK(aB, bB, kc + 16);          // prefetch chunk 2i+1
        COMP_CHUNK(aA, bA);                   // compute  chunk 2i
        const int kn = (kc + 32 < K) ? (kc + 32) : 0;  // wrap: harmless reload
        LOAD_CHUNK(aA, bA, kn);               // prefetch chunk 2i+2
        COMP_CHUNK(aB, bB);                   // compute  chunk 2i+1
    }
#undef LOAD_CHUNK
#undef COMP_CHUNK

    v8f c[4] = {c0, c1, c2, c3};
#pragma unroll
    for (int t = 0; t < 4; ++t) {
#pragma unroll
        for (int r = 0; r < 8; ++r) {
            const int m = mtile + r + 8 * hi;
            if (m < M) {
                const int n = nbase + t * 16 + lo;
                const size_t idx = (size_t)m * N + n;
                float v = alpha * c[t][r];
                if (accumulate)  v += C[idx];
                else if (bias)   v += bias[n];
                C[idx] = v;
            }
        }
    }
}

#if USE_TDM
// ---------------------------------------------------------------------------
// Tensor Data Mover: async-load one 2D tile (80 x 64 fp32) whose rows live at
// stride 1280 floats in global memory, into LDS at `lds_off`, with 2 pad
// dwords appended per 64-dword row (LDS row stride = 66 floats).
// tensor_dim1 = 77 < tile_dim1 = 80: rows 77..79 are OOB -> written as zeros.
// D# packing per CDNA5 ISA. Issued per-wave
// (EXEC ignored); caller gates to one wave and waits on TENSORcnt.
// This toolchain exposes the 6-arg builtin:
//   (uint32x4 g0, int32x8 g1, int32x4 g2, int32x4 g3, int32x8, i32 cpol)
// groups 2/3 and the trailing group are zero for a <=2D tensor.
// ---------------------------------------------------------------------------
__device__ __forceinline__ void tdm_load_tile_77x64(const float* gptr, unsigned lds_off)
{
    const unsigned long long ga = (unsigned long long)(size_t)gptr;

    u32x4 g0;
    g0[0] = 1u;                                              // count=1 (valid), user mode
    g0[1] = lds_off;                                         // LDS byte address
    g0[2] = (unsigned)(ga & 0xFFFFFFFFu);                    // global_addr[31:0]
    g0[3] = (unsigned)((ga >> 32) & 0x01FFFFFFu) | (2u << 30); // global_addr[56:32], type=2

    i32x8 g1;
    g1[0] = (int)((2u << 16)      // data_size = 4 bytes
                | (1u << 20)      // pad_enable
                | (5u << 22)      // pad_interval code 5 -> every 64 dwords
                | (1u << 25));    // pad_amount  code 1 -> 2 dwords
    g1[1] = (int)((64u & 0xFFFFu) << 16);                    // tensor_dim0 = 64 (lo16)
    g1[2] = (int)((77u & 0xFFFFu) << 16);                    // dim0 hi | tensor_dim1 = 77 (lo16)
    g1[3] = (int)(64u << 16);                                // dim1 hi | tile_dim0 = 64
    g1[4] = (int)(80u);                                      // tile_dim1 = 80 | tile_dim2 = 0
    g1[5] = (int)(INNER);                                    // tensor_dim0_stride = 1280 (lo32)
    g1[6] = 0;                                               // stride0 hi | stride1 lo (unused, 2D)
    g1[7] = 0;

    const i32x4 z4 = {0, 0, 0, 0};                           // groups 2/3 unused (<=2D)
    const i32x8 z8 = {0, 0, 0, 0, 0, 0, 0, 0};
    __builtin_amdgcn_tensor_load_to_lds(g0, g1, z4, z4, z8, 0);
}
#endif

// ---------------------------------------------------------------------------
// Fused attention for one (b, h) and a 64-row L chunk (4 waves x 16 rows):
//   scores = (q_bh @ k_bh^T) * 1/sqrt(64)   -> d_out score region
//   prob   = softmax_s(scores)              -> d_out prob region
//   attn   = prob @ v_bh                    -> workspace [B, L, INNER]
//
// K/V tiles are DMA'd into LDS by the Tensor Data Mover (async, TENSORcnt),
// row-major [s][d] with stride 66 floats (TDM LDS padding) -> conflict-free,
// zero-padded to 80 rows by the TDM OOB rule. Score-phase B-frags are single
// aligned ds_load_b64 (k pairs contiguous in d). Probs bounce through LDS to
// re-enter the PV WMMA in A-fragment layout.
// ---------------------------------------------------------------------------
__launch_bounds__(128)
__global__ void attn_softmax_wmma(const float* __restrict__ q,
                                  const float* __restrict__ kb,
                                  const float* __restrict__ vb,
                                  float* __restrict__ score,
                                  float* __restrict__ prob,
                                  float* __restrict__ attn,
                                  int B, int L)
{
    __shared__ float ks[SPAD * KVSTRIDE];   // [s][d], stride 66, rows 77..79 zero
    __shared__ float vs[SPAD * KVSTRIDE];   // [s][d], stride 66, rows 77..79 zero
    __shared__ float ps[4][16 * SPAD];      // per-wave 16 x 80 probabilities

    const int tid  = threadIdx.x;
    const int w    = tid >> 5;
    const int lane = tid & 31;
    const int lo   = lane & 15;
    const int hi   = lane >> 4;
    const int koff = 2 * hi;

    const int bh = blockIdx.y;
    const int b  = bh / HEADS;
    const int h  = bh % HEADS;
    const int m0 = blockIdx.x * 64 + w * 16;

    const float* kp = kb + (size_t)b * SLEN * INNER + h * HEAD_DIM;
    const float* vp = vb + (size_t)b * SLEN * INNER + h * HEAD_DIM;

#if USE_TDM
    if (w == 0) {
        tdm_load_tile_77x64(kp, (unsigned)(size_t)(void*)ks);
        tdm_load_tile_77x64(vp, (unsigned)(size_t)(void*)vs);
        __builtin_amdgcn_s_wait_tensorcnt(0);
    }
#else
    for (int idx = tid; idx < SPAD * HEAD_DIM; idx += 128) {
        const int s = idx >> 6;
        const int d = idx & 63;
        const bool ok = (s < SLEN);
        ks[s * KVSTRIDE + d] = ok ? kp[(size_t)s * INNER + d] : 0.0f;
        vs[s * KVSTRIDE + d] = ok ? vp[(size_t)s * INNER + d] : 0.0f;
    }
#endif
    __syncthreads();

    // ---- scores: 16 x 80 = q(16x64) @ k^T(64x80) ----
    // B-frag element (k=d, n=s) = ks[s*66 + d]; (k, k+1) contiguous -> b64.
    const float* qp = q + ((size_t)b * L + m0 + lo) * INNER + h * HEAD_DIM + koff;
    v8f c[5];
#pragma unroll
    for (int t = 0; t < 5; ++t) c[t] = (v8f)0.0f;

#pragma unroll 4
    for (int k0 = 0; k0 < HEAD_DIM; k0 += 4) {
        const v2f a = *(const v2f*)(qp + k0);
#pragma unroll
        for (int t = 0; t < 5; ++t) {
            const v2f bfr = *(const v2f*)(ks + (t * 16 + lo) * KVSTRIDE + k0 + koff);
            c[t] = WMMA_F32X4(a, bfr, c[t]);
        }
    }

    const float scale = 0.125f;  // 1/sqrt(64)
#pragma unroll
    for (int t = 0; t < 5; ++t) c[t] *= scale;

    // ---- write scaled scores + rowwise softmax (16-lane shuffle reduce) ----
    const size_t srow_base = (size_t)(b * HEADS + h) * L + m0;
#pragma unroll
    for (int r = 0; r < 8; ++r) {
        const int m_local = r + 8 * hi;
        const size_t row_off = (srow_base + m_local) * SLEN;

        float mx = -3.0e38f;
#pragma unroll
        for (int t = 0; t < 5; ++t) {
            const int n = t * 16 + lo;
            if (n < SLEN) {
                const float v = c[t][r];
                score[row_off + n] = v;
                mx = fmaxf(mx, v);
            }
        }
#pragma unroll
        for (int off = 1; off < 16; off <<= 1)
            mx = fmaxf(mx, __shfl_xor(mx, off, 32));

        float e[5];
        float sum = 0.0f;
#pragma unroll
        for (int t = 0; t < 5; ++t) {
            const int n = t * 16 + lo;
            e[t] = (n < SLEN) ? __expf(c[t][r] - mx) : 0.0f;
            sum += e[t];
        }
#pragma unroll
        for (int off = 1; off < 16; off <<= 1)
            sum += __shfl_xor(sum, off, 32);

        const float inv = 1.0f / sum;
#pragma unroll
        for (int t = 0; t < 5; ++t) {
            const int n = t * 16 + lo;
            const float p = e[t] * inv;
            ps[w][m_local * SPAD + n] = p;            // zero-padded cols too
            if (n < SLEN) prob[row_off + n] = p;
        }
    }
    __syncthreads();

    // ---- attn out: 16 x 64 = P(16x80) @ V(80x64) ----
    // A-frag from ps (row-major, contiguous pairs -> b64); B-frag from vs
    // (k = s varies -> two conflict-free b32 loads at stride 66 floats).
    const float* pp = ps[w] + lo * SPAD + koff;
    v8f o0 = (v8f)0.0f, o1 = (v8f)0.0f, o2 = (v8f)0.0f, o3 = (v8f)0.0f;

#pragma unroll 4
    for (int k0 = 0; k0 < SPAD; k0 += 4) {
        const v2f a = *(const v2f*)(pp + k0);
        const float* vr0 = vs + (k0 + koff) * KVSTRIDE;
        const float* vr1 = vr0 + KVSTRIDE;
#pragma unroll
        for (int t = 0; t < 4; ++t) {
            v2f bfr;
            bfr.x = vr0[t * 16 + lo];
            bfr.y = vr1[t * 16 + lo];
            if      (t == 0) o0 = WMMA_F32X4(a, bfr, o0);
            else if (t == 1) o1 = WMMA_F32X4(a, bfr, o1);
            else if (t == 2) o2 = WMMA_F32X4(a, bfr, o2);
            else             o3 = WMMA_F32X4(a, bfr, o3);
        }
    }

    v8f o[4] = {o0, o1, o2, o3};
#pragma unroll
    for (int t = 0; t < 4; ++t) {
#pragma unroll
        for (int r = 0; r < 8; ++r) {
            const int m_local = r + 8 * hi;
            attn[((size_t)b * L + m0 + m_local) * INNER + h * HEAD_DIM + t * 16 + lo] = o[t][r];
        }
    }
}

// ---------------------------------------------------------------------------
// Host-side orchestration. Stream-ordered launches only (graph-capture safe).
// ---------------------------------------------------------------------------
extern "C" void kernel_launch(void* const* d_in, const int* in_sizes, int n_in,
                              void* d_out, int out_size, void* d_ws, size_t ws_size,
                              hipStream_t stream)
{
    (void)in_sizes; (void)n_in; (void)out_size; (void)ws_size;

    const float* hs  = (const float*)d_in[0];   // [8, 4096, 1280]
    const float* enc = (const float*)d_in[1];   // [8, 77, 768]
    const float* Wq  = (const float*)d_in[2];   // [1280, 1280]
    const float* Wk  = (const float*)d_in[3];   // [1280, 768]
    const float* Wv  = (const float*)d_in[4];   // [1280, 768]
    const float* Wo  = (const float*)d_in[5];   // [1280, 1280]
    const float* bo  = (const float*)d_in[6];   // [1280]
    const float* Aq  = (const float*)d_in[7];   // [128, 1280]
    const float* Bq  = (const float*)d_in[8];   // [1280, 128]
    const float* Ak  = (const float*)d_in[9];   // [128, 768]
    const float* Bk  = (const float*)d_in[10];  // [1280, 128]
    const float* Av  = (const float*)d_in[11];  // [128, 768]
    const float* Bv  = (const float*)d_in[12];  // [1280, 128]
    const float* Ao  = (const float*)d_in[13];  // [128, 1280]
    const float* Bo  = (const float*)d_in[14];  // [1280, 128]

    const int B = 8, L = 4096, CTX = 768, R = 128;
    const int Mbig = B * L;        // 32768
    const int Msm  = B * SLEN;     // 616
    const float LS = 16.0f / 128.0f;

    // Workspace carve-up (fp32 elements).
    float* ws   = (float*)d_ws;
    const size_t QN  = (size_t)Mbig * INNER;    // 41,943,040
    const size_t KVN = (size_t)Msm * INNER;     //    788,480
    const size_t XBN = (size_t)Mbig * R;        //  4,194,304
    float* qbuf = ws;
    float* attn = qbuf + QN;
    float* kbuf = attn + QN;
    float* vbuf = kbuf + KVN;
    float* xab  = vbuf + KVN;                   // big LoRA activations
    float* xas  = xab + XBN;                    // small LoRA activations

    // Output carve-up: out | attn_score | attn_prob (flat, fp32).
    float* outp  = (float*)d_out;
    float* score = outp + QN;
    float* prob  = score + (size_t)B * HEADS * L * SLEN;

    const dim3 blk(128);
#define GRID(M, N) dim3((unsigned)((N) / 64), (unsigned)(((M) + 63) / 64))

    // ---- Q projection: q = hs @ Wq^T + LS * (hs @ Aq^T) @ Bq^T ----
    gemm_wmma_f32<<<GRID(Mbig, R),     blk, 0, stream>>>(hs,  Aq, xab,  nullptr, Mbig, R,     INNER, 1.0f, 0);
    gemm_wmma_f32<<<GRID(Mbig, INNER), blk, 0, stream>>>(hs,  Wq, qbuf, nullptr, Mbig, INNER, INNER, 1.0f, 0);
    gemm_wmma_f32<<<GRID(Mbig, INNER), blk, 0, stream>>>(xab, Bq, qbuf, nullptr, Mbig, INNER, R,     LS,   1);

    // ---- K projection (encoder, 616 rows) ----
    gemm_wmma_f32<<<GRID(Msm, R),     blk, 0, stream>>>(enc, Ak, xas,  nullptr, Msm, R,     CTX, 1.0f, 0);
    gemm_wmma_f32<<<GRID(Msm, INNER), blk, 0, stream>>>(enc, Wk, kbuf, nullptr, Msm, INNER, CTX, 1.0f, 0);
    gemm_wmma_f32<<<GRID(Msm, INNER), blk, 0, stream>>>(xas, Bk, kbuf, nullptr, Msm, INNER, R,   LS,   1);

    // ---- V projection ----
    gemm_wmma_f32<<<GRID(Msm, R),     blk, 0, stream>>>(enc, Av, xas,  nullptr, Msm, R,     CTX, 1.0f, 0);
    gemm_wmma_f32<<<GRID(Msm, INNER), blk, 0, stream>>>(enc, Wv, vbuf, nullptr, Msm, INNER, CTX, 1.0f, 0);
    gemm_wmma_f32<<<GRID(Msm, INNER), blk, 0, stream>>>(xas, Bv, vbuf, nullptr, Msm, INNER, R,   LS,   1);

    // ---- Attention (TDM-staged K/V, scores + softmax + prob@V) ----
    attn_softmax_wmma<<<dim3((unsigned)(L / 64), (unsigned)(B * HEADS)), blk, 0, stream>>>(
        qbuf, kbuf, vbuf, score, prob, attn, B, L);

    // ---- Output projection: out = attn @ Wo^T + bo + LS * (attn @ Ao^T) @ Bo^T ----
    gemm_wmma_f32<<<GRID(Mbig, R),     blk, 0, stream>>>(attn, Ao, xab,  nullptr, Mbig, R,     INNER, 1.0f, 0);
    gemm_wmma_f32<<<GRID(Mbig, INNER), blk, 0, stream>>>(attn, Wo, outp, bo,      Mbig, INNER, INNER, 1.0f, 0);
    gemm_wmma_f32<<<GRID(Mbig, INNER), blk, 0, stream>>>(xab,  Bo, outp, nullptr, Mbig, INNER, R,     LS,   1);
#undef GRID
}